// ChainLoss_70497593196987
// MI455X (gfx1250) — compile-verified
//
#include <hip/hip_runtime.h>
#include <stdint.h>

#define NEGV (-1e30f)

// ---------------------------------------------------------------------------
// CDNA5 Tensor Data Mover helpers (gfx1250). D# layout per CDNA5 ISA §8.
// This toolchain exposes the 6-arg builtin:
//   (uint32x4 g0, int32x8 g1, int32x4 g2, int32x4 g3, int32x8 gx, i32 cpol)
// ---------------------------------------------------------------------------
typedef __attribute__((ext_vector_type(4))) unsigned int tdm_u32x4;
typedef __attribute__((ext_vector_type(8))) int          tdm_i32x8;
typedef __attribute__((ext_vector_type(4))) int          tdm_i32x4;

__device__ __forceinline__ void tdm_wait_all() {
#if __has_builtin(__builtin_amdgcn_s_wait_tensorcnt)
  __builtin_amdgcn_s_wait_tensorcnt(0);
#else
  asm volatile("s_wait_tensorcnt 0" ::: "memory");
#endif
}

// Async DMA of `nelem` fp32 (contiguous row) from global -> LDS via TDM.
// 1-row 2D tile: tile_dim0 = nelem, tile_dim1 = 1. Tracked by TENSORcnt.
__device__ __forceinline__ void tdm_load_row(const float* gsrc, float* lds_dst, int nelem) {
  uint64_t ga = (uint64_t)(uintptr_t)gsrc;
  // Low 32 bits of a generic LDS pointer are the LDS byte offset (ISA §10.2).
  unsigned lds_off = (unsigned)(uintptr_t)(void*)lds_dst;

  tdm_u32x4 g0;
  g0.x = 1u;                                                  // count=1 (valid D#)
  g0.y = lds_off;                                             // lds_addr
  g0.z = (unsigned)(ga & 0xFFFFFFFFull);                      // global_addr[31:0]
  g0.w = (unsigned)((ga >> 32) & 0x1FFFFFFull) | 0x80000000u; // addr[56:32] | type=2

  tdm_i32x8 g1;
  g1.s0 = 2 << 16;                          // workgroup_mask=0, data_size=2 (4B)
  g1.s1 = (nelem & 0xFFFF) << 16;           // tensor_dim0[15:0]
  g1.s2 = ((nelem >> 16) & 0xFFFF) | (1 << 16); // tensor_dim0[31:16], tensor_dim1=1
  g1.s3 = (nelem & 0xFFFF) << 16;           // tile_dim0 = nelem
  g1.s4 = 1;                                // tile_dim1 = 1, tile_dim2 = 0
  g1.s5 = nelem;                            // tensor_dim0_stride[31:0]
  g1.s6 = 0;                                // stride0 hi / stride1 lo
  g1.s7 = 0;

  tdm_i32x4 gz4 = (tdm_i32x4)0;             // groups 2/3 unused (<=2D tensor)
  tdm_i32x8 gz8 = (tdm_i32x8)0;             // extra group (6-arg toolchain form)
#if __has_builtin(__builtin_amdgcn_tensor_load_to_lds)
  __builtin_amdgcn_tensor_load_to_lds(g0, g1, gz4, gz4, gz8, 0);
#else
  asm volatile("tensor_load_to_lds %0, %1, %2, %3"
               :: "s"(g0), "s"(g1), "s"(gz4), "s"(gz4) : "memory");
#endif
}

// ---------------------------------------------------------------------------
// LDS float atomic max via monotonic int bit mapping (-> ds_max_i32/ds_min_u32)
// ---------------------------------------------------------------------------
__device__ __forceinline__ void atomic_max_f32_lds(float* addr, float v) {
  int b = __float_as_int(v);
  if (b >= 0) atomicMax((int*)addr, b);
  else        atomicMin((unsigned int*)addr, (unsigned int)b);
}

// ---------------------------------------------------------------------------
// One HMM chain forward per block. per_batch=0: shared graph (denominator),
// per_batch=1: per-utterance graph (numerator). Dynamic LDS layout:
//   alpha[S] | m[S] | s[S] | xbuf[2*D]
// Per-step pipeline (4 barriers/step):
//   wait TDM row t -> issue TDM row t+1 -> seg-max -> seg-sum-exp ->
//   alpha update fused with m/s reset for step t+1
// ---------------------------------------------------------------------------
__global__ __launch_bounds__(1024)
void chain_forward_kernel(const float* __restrict__ x,          // [B,T,D]
                          const int*   __restrict__ x_lengths,  // [B]
                          const int*   __restrict__ src_g,
                          const int*   __restrict__ dst_g,
                          const int*   __restrict__ pdf_g,
                          const float* __restrict__ logw_g,
                          const float* __restrict__ init_g,
                          const float* __restrict__ final_g,
                          int S, int E, int T, int D, int per_batch,
                          float* __restrict__ ll_out) {
  extern __shared__ float smem[];
  float* alpha = smem;
  float* mbuf  = smem + S;
  float* sbuf  = smem + 2 * S;
  float* xbuf  = smem + 3 * S;   // 2*D floats, double buffered

  const int b   = blockIdx.x;
  const int tid = threadIdx.x;
  const int nt  = blockDim.x;
  const int len = x_lengths[b];

  const size_t eo = per_batch ? (size_t)b * (size_t)E : 0;
  const size_t so = per_batch ? (size_t)b * (size_t)S : 0;
  const int*   esrc  = src_g  + eo;
  const int*   edst  = dst_g  + eo;
  const int*   epdf  = pdf_g  + eo;
  const float* elogw = logw_g + eo;
  const float* einit = init_g  + so;
  const float* efin  = final_g + so;
  const float* xb    = x + (size_t)b * (size_t)T * (size_t)D;

  // Prologue: alpha = init, m = NEG, s = 0, and kick off TDM for row 0.
  for (int s = tid; s < S; s += nt) {
    alpha[s] = einit[s];
    mbuf[s]  = NEGV;
    sbuf[s]  = 0.0f;
  }
  const bool leader = (tid < 32);  // wave 0 (gfx1250 is wave32)
  if (leader) tdm_load_row(xb, xbuf, D);
  __syncthreads();

  for (int t = 0; t < len; ++t) {
    const float* cur = xbuf + (size_t)(t & 1) * D;

    if (leader) {
      tdm_wait_all();                             // row t landed in LDS
      if ((t + 1) < len)                          // overlap: DMA row t+1
        tdm_load_row(xb + (size_t)(t + 1) * D, xbuf + (size_t)((t + 1) & 1) * D, D);
    }
    __syncthreads();                              // publish row t

    // pass 1: m[dst] = max over edges of alpha[src] + logw + x_t[pdf]
    for (int e = tid; e < E; e += nt) {
      float sc = alpha[esrc[e]] + elogw[e] + cur[epdf[e]];
      atomic_max_f32_lds(&mbuf[edst[e]], sc);
    }
    __syncthreads();

    // pass 2: s[dst] += exp(score - m[dst])
    for (int e = tid; e < E; e += nt) {
      float sc = alpha[esrc[e]] + elogw[e] + cur[epdf[e]];
      atomicAdd(&sbuf[edst[e]], __expf(sc - mbuf[edst[e]]));
    }
    __syncthreads();

    // alpha = s > 0 ? m + log(s) : NEG, fused with m/s reset for step t+1
    for (int s = tid; s < S; s += nt) {
      float sv = sbuf[s];
      alpha[s] = (sv > 0.0f) ? (mbuf[s] + __logf(sv)) : NEGV;
      mbuf[s]  = NEGV;
      sbuf[s]  = 0.0f;
    }
    __syncthreads();
  }

  // ll = logsumexp(alpha + final); mbuf[0]/sbuf[0] are already NEG/0 here.
  float lmax = NEGV;
  for (int s = tid; s < S; s += nt) lmax = fmaxf(lmax, alpha[s] + efin[s]);
  atomic_max_f32_lds(&mbuf[0], lmax);
  __syncthreads();
  const float gm = mbuf[0];
  float lsum = 0.0f;
  for (int s = tid; s < S; s += nt) lsum += __expf(alpha[s] + efin[s] - gm);
  atomicAdd(&sbuf[0], lsum);
  __syncthreads();
  if (tid == 0) ll_out[b] = gm + __logf(sbuf[0]);
}

// objf = -(sum(num_ll) - sum(den_ll)) / sum(lengths)
__global__ void chain_finalize_kernel(const float* __restrict__ ll_buf,
                                      const int* __restrict__ x_lengths,
                                      int B, float* __restrict__ out) {
  if (blockIdx.x == 0 && threadIdx.x == 0) {
    float den = 0.0f, num = 0.0f; long long lt = 0;
    for (int b = 0; b < B; ++b) {
      den += ll_buf[b];
      num += ll_buf[B + b];
      lt  += (long long)x_lengths[b];
    }
    out[0] = -(num - den) / (float)lt;
  }
}

extern "C" void kernel_launch(void* const* d_in, const int* in_sizes, int n_in,
                              void* d_out, int out_size, void* d_ws, size_t ws_size,
                              hipStream_t stream) {
  const float* x        = (const float*)d_in[0];
  const int*   x_len    = (const int*)  d_in[1];
  const int*   den_src  = (const int*)  d_in[2];
  const int*   den_dst  = (const int*)  d_in[3];
  const int*   den_pdf  = (const int*)  d_in[4];
  const float* den_logw = (const float*)d_in[5];
  const float* den_init = (const float*)d_in[6];
  const float* den_fin  = (const float*)d_in[7];
  const int*   num_src  = (const int*)  d_in[8];
  const int*   num_dst  = (const int*)  d_in[9];
  const int*   num_pdf  = (const int*)  d_in[10];
  const float* num_logw = (const float*)d_in[11];
  const float* num_init = (const float*)d_in[12];
  const float* num_fin  = (const float*)d_in[13];

  const int B     = in_sizes[1];
  const int E_den = in_sizes[2];
  const int S_den = in_sizes[6];
  const int E_num = in_sizes[8]  / B;
  const int S_num = in_sizes[12] / B;
  const int D     = 3072;                       // per reference
  const int T     = in_sizes[0] / (B * D);

  float* ll_ws = (float*)d_ws;                  // [2*B] chain log-likelihoods

  const size_t sh_den = (size_t)(3 * S_den + 2 * D) * sizeof(float); // ~72.5 KB
  const size_t sh_num = (size_t)(3 * S_num + 2 * D) * sizeof(float); // ~27 KB
  (void)hipFuncSetAttribute(reinterpret_cast<const void*>(chain_forward_kernel),
                            hipFuncAttributeMaxDynamicSharedMemorySize,
                            (int)sh_den);

  // Denominator: shared graph, one block (chain) per utterance.
  chain_forward_kernel<<<B, 1024, sh_den, stream>>>(
      x, x_len, den_src, den_dst, den_pdf, den_logw, den_init, den_fin,
      S_den, E_den, T, D, /*per_batch=*/0, ll_ws);

  // Numerator: per-utterance graphs (tiny), one block per utterance.
  chain_forward_kernel<<<B, 256, sh_num, stream>>>(
      x, x_len, num_src, num_dst, num_pdf, num_logw, num_init, num_fin,
      S_num, E_num, T, D, /*per_batch=*/1, ll_ws + B);

  chain_finalize_kernel<<<1, 32, 0, stream>>>(ll_ws, x_len, B, (float*)d_out);
}